// Router_9680856285359
// MI455X (gfx1250) — compile-verified
//
#include <hip/hip_runtime.h>
#include <math.h>

// ---------------------------------------------------------------------------
// MoE top-2 router for MI455X (gfx1250, wave32).
//   logits = x @ w_g^T  via V_WMMA_F32_16X16X4_F32 (fp32 in/out, matches ref)
//   top-2 + softmax-over-top2, then exact sequential capacity assignment via
//   chunked per-expert prefix sums, 335MB output zero-filled with NT b128
//   stores + sparse scatter of the 8192 nonzeros.
// ---------------------------------------------------------------------------

typedef float v2f __attribute__((ext_vector_type(2)));
typedef float v4f __attribute__((ext_vector_type(4)));
typedef float v8f __attribute__((ext_vector_type(8)));

#define NEXP 64
#define TOPK 2
#define NCHUNK 64

// ---- Kernel 1: logits[T,64] = x[T,D] * w_g[64,D]^T --------------------------
// Block = 128 threads = 4 waves. Wave w computes the 16x16 tile
// (rows blockIdx.x*16 .. +15) x (experts w*16 .. +15), accumulating K in
// steps of 4 with V_WMMA_F32_16X16X4_F32.
// A(16x4) lane layout: m = lane&15, VGPR0=K(kb), VGPR1=K(kb+1), kb=(lane>>4)*2
// B(4x16) lane layout: n = lane&15, same K striping.
// C(16x16): VGPR v -> m = v + (lane>>4)*8, n = lane&15.
__global__ __launch_bounds__(128) void router_gemm_wmma(
    const float* __restrict__ x, const float* __restrict__ wg,
    float* __restrict__ logits, int T, int D) {
  const int lane = threadIdx.x & 31;
  const int wave = threadIdx.x >> 5;        // expert tile 0..3
  const int mn   = lane & 15;
  const int kb   = (lane >> 4) << 1;        // 0 or 2
  const int rowTile = blockIdx.x * 16;

  const float* __restrict__ arow = x  + (size_t)(rowTile + mn) * D;
  const float* __restrict__ brow = wg + (size_t)(wave * 16 + mn) * D;

  v8f c = {};
  for (int k = 0; k < D; k += 4) {
    v2f a, b;
    a.x = arow[k + kb];
    a.y = arow[k + kb + 1];
    b.x = brow[k + kb];
    b.y = brow[k + kb + 1];
    // 8 args: (neg_a, A, neg_b, B, c_mod, C, reuse_a, reuse_b)
    c = __builtin_amdgcn_wmma_f32_16x16x4_f32(false, a, false, b,
                                              (short)0, c, false, false);
  }

  const int halo = (lane >> 4) * 8;
  float* outp = logits + (size_t)(rowTile + halo) * NEXP + wave * 16 + mn;
#pragma unroll
  for (int v = 0; v < 8; ++v) outp[(size_t)v * NEXP] = c[v];
}

// ---- Kernel 2: per-token top-2 and softmax over the two logits -------------
// jax.lax.top_k tie-break = lowest index first -> strict '>' comparisons.
__global__ void router_top2(const float* __restrict__ logits,
                            int* __restrict__ top_i, float* __restrict__ top_w,
                            int T) {
  int t = blockIdx.x * blockDim.x + threadIdx.x;
  if (t >= T) return;
  const float* row = logits + (size_t)t * NEXP;
  float v1 = -INFINITY, v2 = -INFINITY;
  int i1 = 0, i2 = 0;
#pragma unroll 4
  for (int e = 0; e < NEXP; ++e) {
    float v = row[e];
    if (v > v1)      { v2 = v1; i2 = i1; v1 = v; i1 = e; }
    else if (v > v2) { v2 = v; i2 = e; }
  }
  // softmax over {v1, v2} (others are -inf in the reference's masked softmax)
  float e2  = __expf(v2 - v1);
  float inv = 1.0f / (1.0f + e2);
  top_i[t * 2 + 0] = i1;
  top_i[t * 2 + 1] = i2;
  top_w[t * 2 + 0] = inv;
  top_w[t * 2 + 1] = e2 * inv;
}

// ---- Kernel 3a: per-chunk, per-expert attempt counts -----------------------
__global__ __launch_bounds__(64) void router_chunk_count(
    const int* __restrict__ top_i, int* __restrict__ counts,
    int total, int chunkSz) {
  int c = blockIdx.x;   // chunk
  int e = threadIdx.x;  // expert
  int beg = c * chunkSz;
  int end = min(beg + chunkSz, total);
  int cnt = 0;
  for (int i = beg; i < end; ++i) cnt += (top_i[i] == e) ? 1 : 0;
  counts[c * NEXP + e] = cnt;
}

// ---- Kernel 3b: exclusive prefix over chunks per expert + used_capacity ----
__global__ __launch_bounds__(64) void router_chunk_prefix(
    const int* __restrict__ counts, int* __restrict__ prefix,
    float* __restrict__ used_out, int nChunks, int cap) {
  int e = threadIdx.x;
  int run = 0;
  for (int c = 0; c < nChunks; ++c) {
    prefix[c * NEXP + e] = run;
    run += counts[c * NEXP + e];
  }
  used_out[e] = (float)min(run, cap);
}

// ---- Kernel 3c: replay each chunk in order, scatter accepted attempts ------
__global__ __launch_bounds__(64) void router_scatter(
    const int* __restrict__ top_i, const float* __restrict__ top_w,
    const int* __restrict__ prefix,
    float* __restrict__ cb, float* __restrict__ mask,
    int total, int chunkSz, int cap) {
  int c = blockIdx.x;
  int e = threadIdx.x;
  int cnt = prefix[c * NEXP + e];
  int beg = c * chunkSz;
  int end = min(beg + chunkSz, total);
  for (int i = beg; i < end; ++i) {
    if (top_i[i] == e) {
      if (cnt < cap) {
        int t = i >> 1;  // TOPK == 2
        size_t off = ((size_t)t * NEXP + e) * (size_t)cap + (size_t)cnt;
        cb[off]   = top_w[i];
        mask[off] = 1.0f;
      }
      ++cnt;
    }
  }
}

// ---- Zero fill: 335MB streaming -> non-temporal 128b stores ----------------
__global__ void zero_fill4(v4f* __restrict__ p, size_t n4) {
  size_t i = (size_t)blockIdx.x * blockDim.x + threadIdx.x;
  if (i < n4) {
    v4f z = {0.f, 0.f, 0.f, 0.f};
    __builtin_nontemporal_store(z, p + i);
  }
}

__global__ void zero_tail(float* __restrict__ p, size_t n) {
  size_t i = (size_t)blockIdx.x * blockDim.x + threadIdx.x;
  if (i < n) p[i] = 0.f;
}

// ---------------------------------------------------------------------------
extern "C" void kernel_launch(void* const* d_in, const int* in_sizes, int n_in,
                              void* d_out, int out_size, void* d_ws,
                              size_t ws_size, hipStream_t stream) {
  const float* x  = (const float*)d_in[0];
  const float* wg = (const float*)d_in[1];

  const int E = NEXP;
  const int D = in_sizes[1] / E;        // 2048
  const int T = in_sizes[0] / D;        // 4096

  // cap = floor(TOP_K * 1.25 * T / E), rounded up to even, min 4
  int cap = (TOPK * 125 * T) / (100 * E);
  cap += cap & 1;
  if (cap < 4) cap = 4;

  const int total   = T * TOPK;                       // 8192 attempts
  const int chunkSz = (total + NCHUNK - 1) / NCHUNK;  // 128

  // workspace layout (all fp32/int32 sized):
  float* logits = (float*)d_ws;                         // T*E
  int*   top_i  = (int*)(logits + (size_t)T * E);       // total
  float* top_w  = (float*)(top_i + total);              // total
  int*   counts = (int*)(top_w + total);                // NCHUNK*E
  int*   prefix = counts + NCHUNK * E;                  // NCHUNK*E

  // output layout: [used_capacity(E)] [cb_weight(T*E*cap)] [sec_mask(T*E*cap)]
  float* used_out = (float*)d_out;
  float* cb       = used_out + E;
  float* mask     = cb + (size_t)T * E * (size_t)cap;

  // 1) zero-fill cb_weight + sec_mask (dominant cost, ~335MB streaming)
  size_t nzero = (size_t)2 * T * E * (size_t)cap;
  size_t n4    = nzero >> 2;
  size_t rem   = nzero & 3;
  if (n4) {
    size_t blocks = (n4 + 255) / 256;
    zero_fill4<<<(unsigned)blocks, 256, 0, stream>>>((v4f*)cb, n4);
  }
  if (rem) {
    zero_tail<<<1, 64, 0, stream>>>(cb + (n4 << 2), rem);
  }

  // 2) logits GEMM via WMMA (T assumed multiple of 16; 4096 is)
  router_gemm_wmma<<<T / 16, 128, 0, stream>>>(x, wg, logits, T, D);

  // 3) top-2 + softmax
  router_top2<<<(T + 255) / 256, 256, 0, stream>>>(logits, top_i, top_w, T);

  // 4) exact sequential capacity assignment (chunked prefix) + scatter
  router_chunk_count<<<NCHUNK, 64, 0, stream>>>(top_i, counts, total, chunkSz);
  router_chunk_prefix<<<1, 64, 0, stream>>>(counts, prefix, used_out, NCHUNK, cap);
  router_scatter<<<NCHUNK, 64, 0, stream>>>(top_i, top_w, prefix, cb, mask,
                                            total, chunkSz, cap);
}